// ODEfuncPortHNN_87789131530317
// MI455X (gfx1250) — compile-verified
//
#include <hip/hip_runtime.h>
#include <math.h>

typedef float v2f __attribute__((ext_vector_type(2)));
typedef float v8f __attribute__((ext_vector_type(8)));

// ---------------------------------------------------------------------------
// Kernel 1: fold C, N, t into a 12x2 weight matrix W (K padded 10 -> 12).
// out0 = dH/dy1 = m . W[:,0]
// out1 = -dH/dy0 + N*dH/dy1 + 0.39*sin(1.4*t) = m . W[:,1]   (sin folded into
// the constant-monomial row, since m[0] == 1).
// Monomial basis (degree<=3): idx(a,b) = d(d+1)/2 + b, d = a+b:
//   [1, y0, y1, y0^2, y0*y1, y1^2, y0^3, y0^2*y1, y0*y1^2, y1^3, 0, 0]
// ---------------------------------------------------------------------------
__global__ void hnn_weights_kernel(const float* __restrict__ t,
                                   const float* __restrict__ C,
                                   const float* __restrict__ N,
                                   float* __restrict__ W) {
  if (threadIdx.x != 0) return;
  const int EA[14] = {1,0, 2,1,0, 3,2,1,0, 4,3,2,1,0};
  const int EB[14] = {0,1, 0,1,2, 0,1,2,3, 0,1,2,3,4};
  float g0[10], g1[10];
  for (int i = 0; i < 10; ++i) { g0[i] = 0.f; g1[i] = 0.f; }
  for (int k = 0; k < 14; ++k) {
    int a = EA[k], b = EB[k];
    float c = C[k];
    if (a > 0) { int d = (a - 1) + b;  g0[d * (d + 1) / 2 + b]       += c * (float)a; }
    if (b > 0) { int d = a + (b - 1);  g1[d * (d + 1) / 2 + (b - 1)] += c * (float)b; }
  }
  float s  = 0.39f * sinf(1.4f * t[0]);
  float nn = N[0];
  for (int i = 0; i < 12; ++i) {
    float w0 = (i < 10) ? g1[i] : 0.f;
    float w1 = (i < 10) ? (-g0[i] + nn * g1[i]) : 0.f;
    W[i * 2 + 0] = w0;
    W[i * 2 + 1] = w1;
  }
  W[0 * 2 + 1] += s;  // constant monomial of column 1
}

// ---------------------------------------------------------------------------
// Kernel 2: streaming  out[Bx2] = monomials(y)[Bx12] @ W[12x2]
// via chained V_WMMA_F32_16X16X4_F32 (3 K-chunks of 4). Memory-bound:
// 64 MB total traffic -> ~2.9 us floor at 23.3 TB/s; WMMA/VALU cost is noise.
// ---------------------------------------------------------------------------
__global__ __launch_bounds__(256) void hnn_main_kernel(const float* __restrict__ y,
                                                       const float* __restrict__ W,
                                                       float* __restrict__ out,
                                                       int n) {
  __shared__ float sbuf[8][64];  // 64 output floats (32 points) per wave
  const int lane = threadIdx.x & 31;
  const int wave = threadIdx.x >> 5;
  const int base = blockIdx.x * 256 + wave * 32;  // first point of this wave

  // ---- Build B fragments (uniform across the loop) -------------------------
  // B 4x16 f32, 2 VGPRs: VGPR0 = {lanes0-15: K0, lanes16-31: K2},
  //                      VGPR1 = {lanes0-15: K1, lanes16-31: K3}.
  // Chunk c covers weight rows 4c..4c+3; only columns 0,1 are non-zero.
  const int ncol  = lane & 15;
  const int khalf = (lane >> 4) << 1;          // 0 for lanes 0-15, 2 for 16-31
  const int ncs   = (ncol < 2) ? ncol : 0;     // safe column index
  const float cm  = (ncol < 2) ? 1.f : 0.f;    // zero out columns >= 2
  v2f b0, b1, b2;
  b0.x = cm * W[(0 + khalf) * 2 + ncs];  b0.y = cm * W[(1 + khalf) * 2 + ncs];
  b1.x = cm * W[(4 + khalf) * 2 + ncs];  b1.y = cm * W[(5 + khalf) * 2 + ncs];
  b2.x = cm * W[(8 + khalf) * 2 + ncs];  b2.y = cm * W[(9 + khalf) * 2 + ncs];

  // ---- Load 32 points, coalesced nontemporal b64 ---------------------------
  const int p  = base + lane;
  const int pc = (p < n) ? p : (n - 1);          // clamp: keep EXEC full for WMMA
  const v2f* yv = reinterpret_cast<const v2f*>(y);
  v2f yy = __builtin_nontemporal_load(yv + pc);
  const float y0 = yy.x, y1 = yy.y;
  const bool hi = (lane >= 16);

  // ---- Two 16-point tiles per wave ----------------------------------------
  #pragma unroll
  for (int h = 0; h < 2; ++h) {
    // Row M = lane&15 of this tile -> fetch that point's coordinates.
    const int src = (lane & 15) + h * 16;
    const float x = __shfl(y0, src, 32);
    const float z = __shfl(y1, src, 32);

    const float x2 = x * x, z2 = z * z, xz = x * z;
    const float m0 = 1.f, m1 = x,      m2 = z,      m3 = x2;
    const float m4 = xz,  m5 = z2,     m6 = x2 * x, m7 = x2 * z;
    const float m8 = x * z2, m9 = z2 * z;            // m10 = m11 = 0

    // A 16x4 f32, 2 VGPRs: lanes0-15 hold K0/K1, lanes16-31 hold K2/K3.
    v2f a0, a1, a2;
    a0.x = hi ? m2 : m0;   a0.y = hi ? m3 : m1;      // K rows 0..3
    a1.x = hi ? m6 : m4;   a1.y = hi ? m7 : m5;      // K rows 4..7
    a2.x = hi ? 0.f : m8;  a2.y = hi ? 0.f : m9;     // K rows 8..11

    v8f acc = {};
    acc = __builtin_amdgcn_wmma_f32_16x16x4_f32(false, a0, false, b0,
                                                (short)0, acc, false, false);
    acc = __builtin_amdgcn_wmma_f32_16x16x4_f32(false, a1, false, b1,
                                                (short)0, acc, false, false);
    acc = __builtin_amdgcn_wmma_f32_16x16x4_f32(false, a2, false, b2,
                                                (short)0, acc, false, false);

    // D[m][n]: VGPR m%8, lane (m/8)*16 + n. Stage columns 0,1 into LDS.
    float* sb = &sbuf[wave][h * 32];
    if (lane < 2) {
      #pragma unroll
      for (int j = 0; j < 8; ++j) sb[2 * j + lane] = acc[j];           // M=j
    } else if (hi && lane < 18) {
      #pragma unroll
      for (int j = 0; j < 8; ++j) sb[2 * (j + 8) + (lane - 16)] = acc[j]; // M=j+8
    }
  }

  __syncthreads();  // order DS stores before the gather-read below

  // ---- Coalesced nontemporal b64 store: lane l writes point base+l --------
  if (p < n) {
    const float* sb = sbuf[wave];
    v2f o;
    o.x = sb[2 * lane + 0];
    o.y = sb[2 * lane + 1];
    __builtin_nontemporal_store(o, reinterpret_cast<v2f*>(out) + p);
  }
}

// ---------------------------------------------------------------------------
extern "C" void kernel_launch(void* const* d_in, const int* in_sizes, int n_in,
                              void* d_out, int out_size, void* d_ws, size_t ws_size,
                              hipStream_t stream) {
  const float* t = (const float*)d_in[0];
  const float* y = (const float*)d_in[1];
  const float* C = (const float*)d_in[2];
  const float* N = (const float*)d_in[3];
  float* out = (float*)d_out;
  float* W   = (float*)d_ws;   // 24 floats

  const int n = in_sizes[1] / 2;  // number of points (4,194,304)

  hnn_weights_kernel<<<1, 32, 0, stream>>>(t, C, N, W);

  const int blocks = (n + 255) / 256;
  hnn_main_kernel<<<blocks, 256, 0, stream>>>(y, W, out, n);
}